// NT_Xent_62319975465314
// MI455X (gfx1250) — compile-verified
//
#include <hip/hip_runtime.h>
#include <hip/hip_bf16.h>
#include <math.h>

// ---------------------------------------------------------------------------
// NT-Xent with summed negatives, MI455X (gfx1250, wave32).
//
// Key algebraic collapse:  sum_m <r_n, mem_m/|mem_m|> = <r_n, s>,
// s = sum_m mem_m/|mem_m|.  Turns the 1024x65536x128 GEMM into a 32MB
// streaming reduction (HBM-bound, ~1.4us at 23.3 TB/s) plus a tiny
// 1024x128 stage, which we express with V_WMMA_F32_16X16X4_F32.
// ---------------------------------------------------------------------------

#define D 128
#define EPS 1e-8f
#define INV_T 10.0f           // 1 / TEMPERATURE
#define RED_BLOCKS 256
#define RED_WAVES 8

typedef float v2f __attribute__((ext_vector_type(2)));
typedef float v8f __attribute__((ext_vector_type(8)));

// ---------------------------------------------------------------------------
// Kernel 1: stream the memory bank, one wave per row (32 lanes x float4 = 128
// floats).  Lane-shuffle reduce |row|^2, accumulate row/|row| into per-lane
// registers, fold waves through LDS, emit deterministic per-block partials.
// ---------------------------------------------------------------------------
__global__ __launch_bounds__(RED_WAVES * 32)
void bank_reduce(const float* __restrict__ mem, float* __restrict__ partials, int M) {
    const int lane = threadIdx.x & 31;
    const int wave = threadIdx.x >> 5;
    const int gw   = blockIdx.x * RED_WAVES + wave;
    const int nw   = gridDim.x * RED_WAVES;

    float a0 = 0.f, a1 = 0.f, a2 = 0.f, a3 = 0.f;

    for (int row = gw; row < M; row += nw) {
        const float4 v = *(const float4*)(mem + (size_t)row * D + lane * 4);
        // prefetch the row this wave will touch next iteration (global_prefetch_b8)
        if (row + nw < M)
            __builtin_prefetch(mem + (size_t)(row + nw) * D + lane * 4, 0, 0);

        float ss = v.x * v.x + v.y * v.y + v.z * v.z + v.w * v.w;
        #pragma unroll
        for (int m = 16; m >= 1; m >>= 1) ss += __shfl_xor(ss, m, 32);

        const float inv = 1.0f / fmaxf(sqrtf(ss), EPS);
        a0 += v.x * inv; a1 += v.y * inv; a2 += v.z * inv; a3 += v.w * inv;
    }

    __shared__ float lds[RED_WAVES * D];
    lds[wave * D + lane * 4 + 0] = a0;
    lds[wave * D + lane * 4 + 1] = a1;
    lds[wave * D + lane * 4 + 2] = a2;
    lds[wave * D + lane * 4 + 3] = a3;
    __syncthreads();

    if (threadIdx.x < D) {
        float t = 0.f;
        #pragma unroll
        for (int w = 0; w < RED_WAVES; w++) t += lds[w * D + threadIdx.x];
        partials[blockIdx.x * D + threadIdx.x] = t;
    }
}

// ---------------------------------------------------------------------------
// Kernel 2: deterministic fold of per-block partials into s[128].
// ---------------------------------------------------------------------------
__global__ void reduce_partials(const float* __restrict__ partials,
                                float* __restrict__ s, int nblocks) {
    const int d = threadIdx.x;               // 128 threads
    float t = 0.f;
    for (int b = 0; b < nblocks; b++) t += partials[b * D + d];
    s[d] = t;
}

// ---------------------------------------------------------------------------
// Kernel 3: one wave per 16-row tile.
//   G1 = RealTile(16x128) @ PertTile^T(128x16)   -> diagonal = raw positives
//   G2 = RealTile(16x128) @ [s s ... s](128x16)  -> any column = raw negatives
// via V_WMMA_F32_16X16X4_F32, 32 K-steps.  Row norms folded in afterwards
// (scaling a row of A scales the same row of D).
//
// f32 16x16x4 fragment layout (ISA 7.12.2):
//   A: lane l -> M = l&15, VGPR0/1 = K = 2*(l>>4), 2*(l>>4)+1
//   B: lane l -> N = l&15, VGPR0/1 = K = 2*(l>>4), 2*(l>>4)+1
//   D: lane l, VGPR v -> row v + 8*(l>>4), col l&15
// ---------------------------------------------------------------------------
__global__ __launch_bounds__(32)
void loss_tiles(const float* __restrict__ real, const float* __restrict__ pert,
                const float* __restrict__ s, float* __restrict__ loss_part) {
    const int lane = threadIdx.x;            // 0..31
    const int base = blockIdx.x * 16;
    const int half = lane >> 4;              // 0: K, K+1   1: K+2, K+3
    const int row  = base + (lane & 15);

    __shared__ float rn[16], pn[16];
    __shared__ float D1[16][16];
    __shared__ float D2[16];

    if (lane < 16) {
        const float* rr = real + (size_t)row * D;
        const float* pp = pert + (size_t)row * D;
        float sr = 0.f, sp = 0.f;
        #pragma unroll
        for (int k = 0; k < D; k += 4) {
            const float4 r4 = *(const float4*)(rr + k);
            const float4 p4 = *(const float4*)(pp + k);
            sr += r4.x * r4.x + r4.y * r4.y + r4.z * r4.z + r4.w * r4.w;
            sp += p4.x * p4.x + p4.y * p4.y + p4.z * p4.z + p4.w * p4.w;
        }
        rn[lane] = fmaxf(sqrtf(sr), EPS);
        pn[lane] = fmaxf(sqrtf(sp), EPS);
    }
    __syncthreads();

    v8f c1 = {};   // real @ pert^T
    v8f c2 = {};   // real @ s (replicated columns)

    #pragma unroll 4
    for (int k = 0; k < D; k += 4) {
        const int off = k + 2 * half;
        const v2f a  = *(const v2f*)(real + (size_t)row * D + off);
        const v2f b1 = *(const v2f*)(pert + (size_t)row * D + off);
        const v2f b2 = *(const v2f*)(s + off);
        c1 = __builtin_amdgcn_wmma_f32_16x16x4_f32(false, a, false, b1,
                                                   (short)0, c1, false, false);
        c2 = __builtin_amdgcn_wmma_f32_16x16x4_f32(false, a, false, b2,
                                                   (short)0, c2, false, false);
    }

    // scatter D fragments through LDS to re-associate rows
    #pragma unroll
    for (int v = 0; v < 8; v++) D1[v + 8 * half][lane & 15] = c1[v];
    if ((lane & 15) == 0) {
        #pragma unroll
        for (int v = 0; v < 8; v++) D2[v + 8 * half] = c2[v];
    }
    __syncthreads();

    float loss = 0.f;
    if (lane < 16) {
        const int i = lane;
        const float pos = D1[i][i] / (rn[i] * pn[i]) * INV_T;
        const float neg = D2[i] / rn[i] * INV_T;
        const float d = neg - pos;                    // loss = log(1 + e^d), stable
        loss = (d > 0.f) ? d + log1pf(expf(-d)) : log1pf(expf(d));
    }
    #pragma unroll
    for (int m = 16; m >= 1; m >>= 1) loss += __shfl_xor(loss, m, 32);
    if (lane == 0) loss_part[blockIdx.x] = loss;
}

// ---------------------------------------------------------------------------
// Kernel 4: deterministic mean.
// ---------------------------------------------------------------------------
__global__ void finalize(const float* __restrict__ loss_part, float* __restrict__ out,
                         int ntiles, int N) {
    if (threadIdx.x == 0) {
        float t = 0.f;
        for (int i = 0; i < ntiles; i++) t += loss_part[i];
        out[0] = t / (float)N;
    }
}

// ---------------------------------------------------------------------------
extern "C" void kernel_launch(void* const* d_in, const int* in_sizes, int n_in,
                              void* d_out, int out_size, void* d_ws, size_t ws_size,
                              hipStream_t stream) {
    const float* real = (const float*)d_in[0];
    const float* pert = (const float*)d_in[1];
    const float* mem  = (const float*)d_in[2];

    const int N = in_sizes[0] / D;   // 1024
    const int M = in_sizes[2] / D;   // 65536
    const int ntiles = N / 16;       // 64

    // workspace: partials[blocks*128] | s[128] | loss_part[ntiles]
    int blocks = RED_BLOCKS;
    const size_t need = (size_t)(blocks * D + D + ntiles) * sizeof(float);
    if (need > ws_size) {
        const long avail = (long)(ws_size / sizeof(float)) - D - ntiles;
        blocks = avail > D ? (int)(avail / D) : 1;
    }

    float* ws        = (float*)d_ws;
    float* partials  = ws;
    float* s_vec     = ws + (size_t)blocks * D;
    float* loss_part = s_vec + D;

    bank_reduce<<<blocks, RED_WAVES * 32, 0, stream>>>(mem, partials, M);
    reduce_partials<<<1, D, 0, stream>>>(partials, s_vec, blocks);
    loss_tiles<<<ntiles, 32, 0, stream>>>(real, pert, s_vec, loss_part);
    finalize<<<1, 32, 0, stream>>>(loss_part, (float*)d_out, ntiles, N);
}